// RecursiveNN_58729382806050
// MI455X (gfx1250) — compile-verified
//
#include <hip/hip_runtime.h>

#define EMBED     512
#define TWO_EMBED 1024
#define NUM_CLASS 5
#define M_TILE    32    // nodes per block
#define CHUNK     256   // K-values staged in LDS per pass (4 passes over K=1024)
#define STR       41    // LDS node-stride: 2*STR mod 64 = 18 -> B-read conflict-free

typedef float v2f __attribute__((ext_vector_type(2)));
typedef float v8f __attribute__((ext_vector_type(8)));

// ---------------------------------------------------------------------------
// 1) Level bucketing: depth(node) = length of left-child chain (exact for the
//    balanced tree). Slot within a level via atomic (order within a level is
//    irrelevant: H values are identical regardless of slot order).
// ---------------------------------------------------------------------------
__global__ void zero_i32_kernel(int* __restrict__ p, int n) {
  int i = blockIdx.x * blockDim.x + threadIdx.x;
  if (i < n) p[i] = 0;
}

__global__ void bucket_kernel(const int* __restrict__ is_leaf,
                              const int* __restrict__ left,
                              int n_nodes, int n_levels, int leaves,
                              int* __restrict__ sched,
                              int* __restrict__ counters) {
  int i = blockIdx.x * blockDim.x + threadIdx.x;
  if (i >= n_nodes) return;
  int d = 0, j = i;
  while (!is_leaf[j]) { j = left[j]; ++d; }          // <=13 hops, parallel
  // off[0]=0 ; off[d] = 2*leaves - 2^(n_levels+1-d)  (perfect tree)
  int off = (d == 0) ? 0 : (2 * leaves - (1 << (n_levels + 1 - d)));
  int slot = atomicAdd(&counters[d], 1);
  sched[off + slot] = i;
}

// ---------------------------------------------------------------------------
// 2) Leaves: H[node] = relu(emb[word[node]])   (128 threads x float4 = 512)
// ---------------------------------------------------------------------------
__global__ void leaf_kernel(const int* __restrict__ is_leaf,
                            const int* __restrict__ word,
                            const float* __restrict__ emb,
                            float* __restrict__ H) {
  int node = blockIdx.x;
  if (!is_leaf[node]) return;
  const float4* src = (const float4*)(emb + (size_t)word[node] * EMBED);
  float4* dst = (float4*)(H + (size_t)node * EMBED);
  float4 v = src[threadIdx.x];
  v.x = fmaxf(v.x, 0.f); v.y = fmaxf(v.y, 0.f);
  v.z = fmaxf(v.z, 0.f); v.w = fmaxf(v.w, 0.f);
  dst[threadIdx.x] = v;
}

// ---------------------------------------------------------------------------
// 3) One tree level: Out^T[512, M] = W[512,1024] x Pair^T[1024, M], relu+bias.
//    Block = 32 nodes x 512 features. 8 waves; each wave owns 4 feature tiles
//    (64 features) x 2 node tiles (32 nodes) = 8 accumulators; one A (W)
//    fragment feeds both node tiles -> 0.5 global loads per WMMA.
//    Pair^T staged in LDS in four 256-K chunks (each chunk = one child half).
//    WMMA: V_WMMA_F32_16X16X4_F32 (native fp32: 13-deep matmul chain keeps
//    full fp32 fidelity vs the fp32 reference).
// ---------------------------------------------------------------------------
__global__ __launch_bounds__(256)
void level_gemm_kernel(float* H,
                       const float* __restrict__ W,
                       const float* __restrict__ bW,
                       const int* __restrict__ sched,
                       const int* __restrict__ left,
                       const int* __restrict__ right,
                       int level_off, int level_count) {
  __shared__ float s_pairT[CHUNK * STR];            // 256 x 41 fp32 = 42KB
  __shared__ int s_node[M_TILE], s_left[M_TILE], s_right[M_TILE];

  const int tid = threadIdx.x;
  const int tileBase = blockIdx.x * M_TILE;

  if (tid < M_TILE) {
    int m = tileBase + tid;
    int node = (m < level_count) ? sched[level_off + m] : -1;
    s_node[tid]  = node;
    s_left[tid]  = (node >= 0) ? left[node]  : 0;
    s_right[tid] = (node >= 0) ? right[node] : 0;
  }

  const int wave   = tid >> 5;
  const int lane   = tid & 31;
  const int m_lane = lane & 15;            // B/C/D: lane -> N (node in tile)
  const int koff   = (lane >> 4) << 1;     // A/B: upper half-wave -> K+2
  const int nBase  = wave * 64;            // 4 feature tiles per wave

  // A operand row base per feature tile (row of W), per ISA A-layout lane->M
  int wrow[4];
#pragma unroll
  for (int f = 0; f < 4; ++f)
    wrow[f] = (nBase + 16 * f + m_lane) * TWO_EMBED;

  const v8f vz = {0.f, 0.f, 0.f, 0.f, 0.f, 0.f, 0.f, 0.f};
  v8f acc[4][2] = {{vz, vz}, {vz, vz}, {vz, vz}, {vz, vz}};

  for (int chunk = 0; chunk < 4; ++chunk) {
    const int kGlobalBase = chunk * CHUNK;
    __syncthreads();
    // ---- stage one 256-K chunk of Pair^T (transposed) into LDS ----
    // chunk 0,1 -> left child rows; chunk 2,3 -> right child rows
    for (int c = tid; c < M_TILE * (CHUNK / 4); c += 256) {
      int m    = c >> 6;                   // 64 float4 per node per chunk
      int k4   = c & 63;
      int kloc = k4 << 2;
      int kg   = kGlobalBase + kloc;
      float4 v = make_float4(0.f, 0.f, 0.f, 0.f);
      if (s_node[m] >= 0) {
        int child = (kg < EMBED) ? s_left[m] : s_right[m];
        int kc    = kg & (EMBED - 1);
        v = *(const float4*)(H + (size_t)child * EMBED + kc);  // coalesced
      }
      float* d = &s_pairT[kloc * STR + m];
      d[0 * STR] = v.x; d[1 * STR] = v.y;
      d[2 * STR] = v.z; d[3 * STR] = v.w;
    }
    __syncthreads();

    const float* wb = W + kGlobalBase;
    // ---- K loop: 64 steps of K=4 per chunk; 8 WMMAs per step ----
    for (int k0 = 0; k0 < CHUNK; k0 += 4) {
      int kk = k0 + koff;
      v2f b0, b1;
      b0.x = s_pairT[kk * STR + m_lane];
      b0.y = s_pairT[(kk + 1) * STR + m_lane];
      b1.x = s_pairT[kk * STR + 16 + m_lane];
      b1.y = s_pairT[(kk + 1) * STR + 16 + m_lane];
#pragma unroll
      for (int f = 0; f < 4; ++f) {
        v2f a = *(const v2f*)(wb + wrow[f] + kk);
        acc[f][0] = __builtin_amdgcn_wmma_f32_16x16x4_f32(
            false, a, false, b0, (short)0, acc[f][0], false, false);
        acc[f][1] = __builtin_amdgcn_wmma_f32_16x16x4_f32(
            false, a, false, b1, (short)0, acc[f][1], false, false);
      }
    }
  }

  // ---- epilogue: C/D layout lane->node, VGPR r -> feature r (+8 upper) ----
#pragma unroll
  for (int nt = 0; nt < 2; ++nt) {
    int node = s_node[16 * nt + m_lane];
    if (node < 0) continue;
#pragma unroll
    for (int f = 0; f < 4; ++f) {
      int fBase = nBase + 16 * f + ((lane >> 4) << 3);
      float4 b0 = *(const float4*)(bW + fBase);
      float4 b1 = *(const float4*)(bW + fBase + 4);
      v8f r = acc[f][nt];
      float4 o0, o1;
      o0.x = fmaxf(r[0] + b0.x, 0.f); o0.y = fmaxf(r[1] + b0.y, 0.f);
      o0.z = fmaxf(r[2] + b0.z, 0.f); o0.w = fmaxf(r[3] + b0.w, 0.f);
      o1.x = fmaxf(r[4] + b1.x, 0.f); o1.y = fmaxf(r[5] + b1.y, 0.f);
      o1.z = fmaxf(r[6] + b1.z, 0.f); o1.w = fmaxf(r[7] + b1.w, 0.f);
      float* dst = H + (size_t)node * EMBED + fBase;
      *(float4*)(dst)     = o0;
      *(float4*)(dst + 4) = o1;
    }
  }
}

// ---------------------------------------------------------------------------
// 4) Projection: out[node, c] = H[node] . P[c] + bP[c]   (one wave per node)
// ---------------------------------------------------------------------------
__global__ void proj_kernel(const float* __restrict__ H,
                            const float* __restrict__ P,
                            const float* __restrict__ bP,
                            float* __restrict__ out) {
  int node = blockIdx.x;
  int lane = threadIdx.x;  // 32
  const float* h = H + (size_t)node * EMBED;
  float hv[16];
#pragma unroll
  for (int i = 0; i < 16; ++i) hv[i] = h[lane + 32 * i];
#pragma unroll
  for (int c = 0; c < NUM_CLASS; ++c) {
    const float* p = P + (size_t)c * EMBED;
    float s = 0.f;
#pragma unroll
    for (int i = 0; i < 16; ++i) s += hv[i] * p[lane + 32 * i];
    for (int o = 16; o > 0; o >>= 1) s += __shfl_xor(s, o, 32);
    if (lane == 0) out[(size_t)node * NUM_CLASS + c] = s + bP[c];
  }
}

// ---------------------------------------------------------------------------
extern "C" void kernel_launch(void* const* d_in, const int* in_sizes, int n_in,
                              void* d_out, int out_size, void* d_ws, size_t ws_size,
                              hipStream_t stream) {
  const int*   is_leaf = (const int*)  d_in[0];
  const int*   word    = (const int*)  d_in[1];
  const int*   left    = (const int*)  d_in[2];
  const int*   right   = (const int*)  d_in[3];
  const float* emb     = (const float*)d_in[4];
  const float* W       = (const float*)d_in[5];
  const float* bW      = (const float*)d_in[6];
  const float* P       = (const float*)d_in[7];
  const float* bP      = (const float*)d_in[8];

  const int n_nodes = in_sizes[0];            // 16383
  const int leaves  = (n_nodes + 1) / 2;      // 8192 (perfect tree)
  int n_levels = 0;
  while ((1 << n_levels) < leaves) ++n_levels;  // 13

  // workspace layout: [counters(pad to 256B)] [sched n_nodes ints] [H]
  char* ws = (char*)d_ws;
  int* counters = (int*)ws;
  int* sched    = (int*)(ws + 256);
  size_t schedBytes = ((size_t)n_nodes * sizeof(int) + 255) & ~(size_t)255;
  float* H = (float*)(ws + 256 + schedBytes);  // n_nodes*512 fp32 = 33.5 MB

  zero_i32_kernel<<<1, 32, 0, stream>>>(counters, 32);
  bucket_kernel<<<(n_nodes + 255) / 256, 256, 0, stream>>>(
      is_leaf, left, n_nodes, n_levels, leaves, sched, counters);
  leaf_kernel<<<n_nodes, EMBED / 4, 0, stream>>>(is_leaf, word, emb, H);

  // one GEMM per level (stream order enforces child-before-parent)
  for (int d = 1; d <= n_levels; ++d) {
    int cnt = leaves >> d;
    int off = 2 * leaves - (1 << (n_levels + 1 - d));
    int tiles = (cnt + M_TILE - 1) / M_TILE;
    level_gemm_kernel<<<tiles, 256, 0, stream>>>(H, W, bW, sched, left, right,
                                                 off, cnt);
  }

  proj_kernel<<<n_nodes, 32, 0, stream>>>(H, P, bP, (float*)d_out);
}